// Nbp_51994874085630
// MI455X (gfx1250) — compile-verified
//
#include <hip/hip_runtime.h>
#include <cstdint>

// ---------------------------------------------------------------------------
// Neural BP decoder (B=8, M=384, N=1536, L=20, ROW_W=8) for MI455X / gfx1250.
// Sparse reformulation: msg_dn is supported exactly on H's 3072 nonzeros.
// One workgroup per batch, all state LDS-resident, deterministic CSC gathers,
// double-buffered GLOBAL_LOAD_ASYNC_TO_LDS_B128 prefetch of per-layer weights.
// ---------------------------------------------------------------------------

namespace {
constexpr int kB   = 8;
constexpr int kM   = 384;
constexpr int kN   = 1536;
constexpr int kL   = 20;
constexpr int kRW  = 8;
constexpr int kNNZ = kM * kRW;     // 3072 nonzeros
constexpr int kMaxDeg  = 32;       // column-degree capacity (mean degree = 2)
constexpr int kThreads = 384;      // 12 wave32 waves; 1 thread per check row

// workspace byte offsets (all 16B aligned for b128 async sources)
constexpr size_t kOffCols    = 0;                                  // int[kNNZ]
constexpr size_t kOffColCnt  = kOffCols   + sizeof(int) * kNNZ;    // int[kN]
constexpr size_t kOffColRows = kOffColCnt + sizeof(int) * kN;      // int[kN*kMaxDeg]
constexpr size_t kOffWdeG    = kOffColRows + sizeof(int) * kN * kMaxDeg; // float[kL*kNNZ]
constexpr size_t kOffMargG   = kOffWdeG   + sizeof(float) * kL * kNNZ;   // float[kNNZ]
constexpr size_t kOffLosses  = kOffMargG  + sizeof(float) * kNNZ;        // float[kB]
}  // namespace

// ---- CDNA5 async-copy-to-LDS helpers (ASYNCcnt path, CDNA5 ISA) --------
__device__ __forceinline__ void nbp_async_b128(uint32_t lds_byte, uint64_t gaddr) {
#if defined(__gfx1250__)
  // per-lane copy of 16 bytes from the global address in vaddr to the LDS address in vdst; tracked by ASYNCcnt
  asm volatile("global_load_async_to_lds_b128 %0, %1, off"
               :: "v"(lds_byte), "v"(gaddr)
               : "memory");
#else
  (void)lds_byte; (void)gaddr;
#endif
}

__device__ __forceinline__ void nbp_wait_async0() {
#if defined(__gfx1250__)
  asm volatile("s_wait_asynccnt 0" ::: "memory");
#endif
}

// low 32 bits of a generic pointer to __shared__ data == LDS byte address
__device__ __forceinline__ uint32_t nbp_lds_off(const void* p) {
  return (uint32_t)(uintptr_t)p;
}

// ---------------------------------------------------------------------------
// (1) Extract the kRW column indices of each row of dense H.
__global__ void nbp_build_rows(const float* __restrict__ H, int* __restrict__ cols) {
  int m = blockIdx.x * blockDim.x + threadIdx.x;
  if (m >= kM) return;
  int cnt = 0;
  for (int n = 0; n < kN; ++n) {
    if (H[(size_t)m * kN + n] != 0.0f && cnt < kRW) cols[m * kRW + cnt++] = n;
  }
}

// (2) Deterministic CSC: per column, contributor nonzero-entry ids in
//     ascending order (fixed float summation order -> bit-stable output).
__global__ void nbp_build_csc(const int* __restrict__ cols,
                              int* __restrict__ colcnt, int* __restrict__ colrows) {
  int n = blockIdx.x * blockDim.x + threadIdx.x;
  if (n >= kN) return;
  int cnt = 0;
  for (int e = 0; e < kNNZ; ++e) {
    if (cols[e] == n && cnt < kMaxDeg) colrows[n * kMaxDeg + cnt++] = e;
  }
  colcnt[n] = cnt;
}

// (3) Gather per-layer w_de and marg_weights_de onto the nonzero pattern
//     (dense 12KB/layer tiles -> contiguous 16B-aligned async-copy sources).
__global__ void nbp_gather(const float* __restrict__ wde, const float* __restrict__ marg_de,
                           const int* __restrict__ cols,
                           float* __restrict__ wde_g, float* __restrict__ marg_g) {
  int idx = blockIdx.x * blockDim.x + threadIdx.x;
  if (idx < kL * kNNZ) {
    int e = idx % kNNZ;
    int l = idx / kNNZ;
    int m = e / kRW;
    wde_g[idx] = wde[(size_t)l * kM * kN + (size_t)m * kN + cols[e]];
  }
  if (idx < kNNZ) {
    int m = idx / kRW;
    marg_g[idx] = marg_de[(size_t)m * kN + cols[idx]];
  }
}

// ---------------------------------------------------------------------------
// (4) Decode: one workgroup per batch, 20 layers, 3 barriers/layer,
//     async double-buffered per-layer weight prefetch.
__global__ __launch_bounds__(kThreads) void nbp_decode(
    const float* __restrict__ synd,     // (B,M,1)
    const float* __restrict__ errors,   // (B,N)
    const float* __restrict__ llrs,     // (N)
    const float* __restrict__ wllr,     // (L,N)  async source
    const float* __restrict__ margllr,  // (N)
    const float* __restrict__ resw,     // (L)
    const float* __restrict__ rhos,     // (L)
    const int*   __restrict__ cols,     // (NNZ) row -> column ids
    const int*   __restrict__ colcnt,   // (N)
    const int*   __restrict__ colrows,  // (N,kMaxDeg) column -> entry ids
    const float* __restrict__ wde_g,    // (L,NNZ) async source
    const float* __restrict__ marg_g,   // (NNZ)
    float* __restrict__ losses) {       // (B)
  __shared__ __align__(16) float s_wde[2][kNNZ];   // 2 x 12 KB
  __shared__ __align__(16) float s_wllr[2][kN];    // 2 x 6 KB
  __shared__ float s_msg[kNNZ];                    // 12 KB
  __shared__ float s_acc[kN];                      // 6 KB
  __shared__ float s_red[kThreads];                // 1.5 KB

  const int b   = blockIdx.x;
  const int tid = threadIdx.x;
  const int m   = tid;  // owned check row

  // ---- row-local caches -------------------------------------------------
  int   c[kRW];
  float llc[kRW], msgr[kRW];
  #pragma unroll
  for (int j = 0; j < kRW; ++j) {
    c[j]   = cols[m * kRW + j];
    llc[j] = llrs[c[j]];
    msgr[j] = 0.0f;
    s_msg[m * kRW + j] = 0.0f;
  }
  const float sgn = 1.0f - 2.0f * synd[b * kM + m];

  // ---- column-local caches (4 strided columns per thread) ---------------
  float bb[4], em[4];
  int   cc[4];
  #pragma unroll
  for (int k = 0; k < 4; ++k) {
    int n = tid + k * kThreads;
    bb[k] = llrs[n] * margllr[n];
    em[k] = 1.0f - errors[b * kN + n];
    cc[k] = colcnt[n];
  }

  // ---- async-load layer-0 weights into buffer 0 --------------------------
  {
    uint64_t s0 = (uint64_t)(uintptr_t)wde_g;                  // layer 0 tile
    uint32_t d0 = nbp_lds_off(&s_wde[0][0]);
    nbp_async_b128(d0 + (uint32_t)tid * 16u,              s0 + (uint64_t)tid * 16u);
    nbp_async_b128(d0 + (uint32_t)(tid + kThreads) * 16u, s0 + (uint64_t)(tid + kThreads) * 16u);
    uint64_t s1 = (uint64_t)(uintptr_t)wllr;
    uint32_t d1 = nbp_lds_off(&s_wllr[0][0]);
    nbp_async_b128(d1 + (uint32_t)tid * 16u, s1 + (uint64_t)tid * 16u);
  }
  nbp_wait_async0();
  __syncthreads();

  float lp = 0.0f;

  for (int l = 0; l < kL; ++l) {
    const int buf = l & 1;

    // prefetch layer l+1 into the other buffer (its last reads finished
    // behind the previous layer-end barrier)
    if (l + 1 < kL) {
      uint64_t s0 = (uint64_t)(uintptr_t)(wde_g + (size_t)(l + 1) * kNNZ);
      uint32_t d0 = nbp_lds_off(&s_wde[buf ^ 1][0]);
      nbp_async_b128(d0 + (uint32_t)tid * 16u,              s0 + (uint64_t)tid * 16u);
      nbp_async_b128(d0 + (uint32_t)(tid + kThreads) * 16u, s0 + (uint64_t)(tid + kThreads) * 16u);
      uint64_t s1 = (uint64_t)(uintptr_t)(wllr + (size_t)(l + 1) * kN);
      uint32_t d1 = nbp_lds_off(&s_wllr[buf ^ 1][0]);
      nbp_async_b128(d1 + (uint32_t)tid * 16u, s1 + (uint64_t)tid * 16u);
    }

    const float rw  = resw[l];
    const float rho = rhos[l];

    // (A) column sums: acc[n] = sum_m msg[m,n] * w_de[l,m,n]  (deterministic order)
    #pragma unroll
    for (int k = 0; k < 4; ++k) {
      int n = tid + k * kThreads;
      int base = n * kMaxDeg;
      float s = 0.0f;
      for (int i = 0; i < cc[k]; ++i) {
        int e = colrows[base + i];
        s += s_msg[e] * s_wde[buf][e];
      }
      s_acc[n] = s;
    }
    __syncthreads();

    // (B) check-row update: tanh product with self-exclusion by division
    float d[kRW];
    float p = 1.0f;
    #pragma unroll
    for (int j = 0; j < kRW; ++j) {
      float en = llc[j] * s_wllr[buf][c[j]] + s_acc[c[j]] - msgr[j];
      float t  = tanhf(0.5f * en);
      if (t == 0.0f) t = 1.0f;   // matches jnp.where(d == 0, 1, d)
      d[j] = t;
      p *= t;
    }
    #pragma unroll
    for (int j = 0; j < kRW; ++j) {
      float nm = 2.0f * atanhf(p / d[j]) * sgn + rw * msgr[j];
      msgr[j] = nm;
      s_msg[m * kRW + j] = nm;
    }
    __syncthreads();

    // (C) beliefs + per-layer loss contribution (summed over n, scaled by rho)
    float part = 0.0f;
    #pragma unroll
    for (int k = 0; k < 4; ++k) {
      int n = tid + k * kThreads;
      int base = n * kMaxDeg;
      float s = 0.0f;
      for (int i = 0; i < cc[k]; ++i) {
        int e = colrows[base + i];
        s += s_msg[e] * marg_g[e];
      }
      float bel = s + bb[k];
      // softplus(bel) - (1 - err) * bel, overflow-safe softplus
      part += (fmaxf(bel, 0.0f) + log1pf(expf(-fabsf(bel)))) - em[k] * bel;
    }
    lp += rho * part;

    nbp_wait_async0();   // next layer's weights landed in LDS
    __syncthreads();     // ...and are visible; s_msg reads done before reuse
  }

  // ---- deterministic block reduction of per-thread loss partials ---------
  s_red[tid] = lp;
  __syncthreads();
  for (int s = kThreads / 2; s >= 6; s >>= 1) {   // 384 -> 192 -> ... -> 6
    if (tid < s) s_red[tid] += s_red[tid + s];
    __syncthreads();
  }
  if (tid == 0) {
    float t = s_red[0] + s_red[1] + s_red[2] + s_red[3] + s_red[4] + s_red[5];
    losses[b] = t;
  }
}

// (5) mean over batch
__global__ void nbp_finalize(const float* __restrict__ losses, float* __restrict__ out) {
  float s = 0.0f;
  for (int b = 0; b < kB; ++b) s += losses[b];
  out[0] = s / (float)kB;
}

// ---------------------------------------------------------------------------
extern "C" void kernel_launch(void* const* d_in, const int* in_sizes, int n_in,
                              void* d_out, int out_size, void* d_ws, size_t ws_size,
                              hipStream_t stream) {
  (void)in_sizes; (void)n_in; (void)out_size; (void)ws_size;

  const float* synd    = (const float*)d_in[0];  // (B,M,1)
  const float* errors  = (const float*)d_in[1];  // (B,N)
  const float* H       = (const float*)d_in[2];  // (M,N)
  const float* llrs    = (const float*)d_in[3];  // (N)
  const float* wde     = (const float*)d_in[4];  // (L,M,N)
  const float* wllr    = (const float*)d_in[5];  // (L,N)
  const float* margde  = (const float*)d_in[6];  // (M,N)
  const float* margllr = (const float*)d_in[7];  // (N)
  const float* resw    = (const float*)d_in[8];  // (L)
  const float* rhos    = (const float*)d_in[9];  // (L)
  float* out = (float*)d_out;

  char* ws = (char*)d_ws;
  int*   cols    = (int*)(ws + kOffCols);
  int*   colcnt  = (int*)(ws + kOffColCnt);
  int*   colrows = (int*)(ws + kOffColRows);
  float* wde_g   = (float*)(ws + kOffWdeG);
  float* marg_g  = (float*)(ws + kOffMargG);
  float* losses  = (float*)(ws + kOffLosses);

  nbp_build_rows<<<(kM + 127) / 128, 128, 0, stream>>>(H, cols);
  nbp_build_csc<<<(kN + 127) / 128, 128, 0, stream>>>(cols, colcnt, colrows);
  nbp_gather<<<(kL * kNNZ + 255) / 256, 256, 0, stream>>>(wde, margde, cols, wde_g, marg_g);
  nbp_decode<<<kB, kThreads, 0, stream>>>(synd, errors, llrs, wllr, margllr, resw, rhos,
                                          cols, colcnt, colrows, wde_g, marg_g, losses);
  nbp_finalize<<<1, 1, 0, stream>>>(losses, out);
}